// dmc_68762426409235
// MI455X (gfx1250) — compile-verified
//
#include <hip/hip_runtime.h>

// Muskingum-Cunge river routing on a random tree DAG (N=4096 reaches, T=730 steps).
// adj has exactly one nonzero per column (parent pointer ds[j] > j), so:
//   * inflow  = scatter-add over tree
//   * A-solve = forward substitution by tree level (depth ~25)
// Pipeline: coeff kernel -> WMMA adjacency reduction (ds extraction) ->
//           single persistent workgroup doing all 730 steps in LDS.

#define NR 4096
#define NT 730
#define DT_S 3600.0f
#define VEL_LB_C 0.3f
#define VEL_UB_C 15.0f
#define Q_LB_C 1e-4f
#define SLOPE_LB_C 1e-4f
#define RT_THREADS 256
#define MAXL 1024

typedef float v2f __attribute__((ext_vector_type(2)));
typedef float v8f __attribute__((ext_vector_type(8)));

// ---------------------------------------------------------------------------
// Kernel 1: per-reach Muskingum coefficients c1..c4 (O(N), trivial)
// ---------------------------------------------------------------------------
__global__ void __launch_bounds__(256) mc_coeff_kernel(
    const float* __restrict__ nman, const float* __restrict__ qsp,
    const float* __restrict__ psp,  const float* __restrict__ len,
    const float* __restrict__ slp,  const float* __restrict__ wid,
    const float* __restrict__ xst,
    float* __restrict__ c1, float* __restrict__ c2,
    float* __restrict__ c3, float* __restrict__ c4) {
  int i = blockIdx.x * blockDim.x + threadIdx.x;
  if (i >= NR) return;
  float s   = fmaxf(slp[i], SLOPE_LB_C);
  float dep = logf(wid[i] / psp[i]) / logf(qsp[i]);
  float v   = (1.0f / nman[i]) * powf(dep, 2.0f / 3.0f) * sqrtf(s);
  float c   = fminf(fmaxf(v, VEL_LB_C), VEL_UB_C) * (5.0f / 3.0f);
  float k   = len[i] / c;
  float x   = xst[i];
  float den = 2.0f * k * (1.0f - x) + DT_S;
  c1[i] = (DT_S - 2.0f * k * x) / den;
  c2[i] = (DT_S + 2.0f * k * x) / den;
  c3[i] = (2.0f * k * (1.0f - x) - DT_S) / den;
  c4[i] = (2.0f * DT_S) / den;
}

// ---------------------------------------------------------------------------
// Kernel 2: ds[j] = sum_i i * adj[i][j] via V_WMMA_F32_16X16X4_F32.
// One wave per 16-column block; A holds row indices in matrix row M=0 only
// (per ISA 16x4 f32 A layout: lanes 0-15 = M, VGPR0=K(2*half), VGPR1=K+1).
// Exactly one nonzero per column => fp32 sum is exact. 4 independent
// accumulators for ILP; D tiles add elementwise.
// ---------------------------------------------------------------------------
__global__ void __launch_bounds__(32) mc_ds_wmma_kernel(
    const float* __restrict__ adj, int* __restrict__ dsg) {
  const int j0   = blockIdx.x << 4;   // 16 columns per wave
  const int lane = (int)threadIdx.x;  // 0..31, wave32
  const int n    = lane & 15;
  const int half = lane >> 4;
  v8f acc[4] = {};
  for (int i0 = 0; i0 < NR; i0 += 16) {
#pragma unroll
    for (int c = 0; c < 4; ++c) {
      const int r0 = i0 + 4 * c + 2 * half;  // this lane's K rows: r0, r0+1
      v2f a, b;
      a[0] = (n == 0) ? (float)(r0)     : 0.0f;  // A[M=0][K]
      a[1] = (n == 0) ? (float)(r0 + 1) : 0.0f;
      b[0] = adj[(size_t)r0 * NR + (size_t)(j0 + n)];        // B[K][N=n]
      b[1] = adj[(size_t)(r0 + 1) * NR + (size_t)(j0 + n)];
      acc[c] = __builtin_amdgcn_wmma_f32_16x16x4_f32(
          false, a, false, b, (short)0, acc[c], false, false);
    }
  }
  v8f tot = acc[0] + acc[1] + acc[2] + acc[3];
  // D layout: VGPR0, lanes 0-15 hold M=0, N=lane
  if (lane < 16) dsg[j0 + lane] = (int)(tot[0] + 0.5f);
}

// ---------------------------------------------------------------------------
// Kernel 3: persistent single-workgroup routing. State in LDS:
//   s_q (16KB), s_acc (16KB, doubles as int depth scratch), s_ds (16KB),
//   level table (~8KB)  => ~56KB, well under the 320KB WGP budget.
// Per step: phase1 accum init, phase2 inflow scatter (ds_add_f32 atomics),
// phase3 forward substitution by level (~25 barriers), emit outlet sample.
// ---------------------------------------------------------------------------
__global__ void __launch_bounds__(RT_THREADS) mc_route_kernel(
    const float* __restrict__ q_prime, const int* __restrict__ dsg,
    int* __restrict__ orderg,
    const float* __restrict__ c1g, const float* __restrict__ c2g,
    const float* __restrict__ c3g, const float* __restrict__ c4g,
    float* __restrict__ out) {
  __shared__ float s_q[NR];
  __shared__ float s_acc[NR];
  __shared__ int   s_ds[NR];
  __shared__ int   s_lvl[MAXL + 1];
  __shared__ int   s_pos[MAXL];
  __shared__ int   s_nlev;

  const int tid = (int)threadIdx.x;

  for (int i = tid; i < NR; i += RT_THREADS) {
    s_ds[i] = dsg[i];
    s_q[i]  = q_prime[i];   // q0 = q_prime[0] (NOT clipped, per reference)
  }
  __syncthreads();

  if (tid == 0) {
    out[0] = fmaxf(s_q[NR - 1], Q_LB_C);   // gage at outlet, t=0
    // Tree levels: children of i all have index < i => one ascending pass.
    int* depth = (int*)s_acc;              // reuse accum LDS as int scratch
    for (int j = 0; j < NR; ++j) depth[j] = 0;
    for (int j = 0; j < NR - 1; ++j) {
      int d = depth[j] + 1;
      int p = s_ds[j];
      if (d > depth[p]) depth[p] = d;
    }
    int maxd = 0;
    for (int j = 0; j < NR; ++j) if (depth[j] > maxd) maxd = depth[j];
    if (maxd > MAXL - 1) maxd = MAXL - 1;  // safety clamp (E[maxd] ~ 25)
    int nlev = maxd + 1;
    s_nlev = nlev;
    for (int d = 0; d <= nlev; ++d) s_lvl[d] = 0;
    for (int j = 0; j < NR; ++j) {
      int d = depth[j]; if (d > maxd) d = maxd;
      s_lvl[d + 1]++;
    }
    for (int d = 0; d < nlev; ++d) s_lvl[d + 1] += s_lvl[d];
    for (int d = 0; d < nlev; ++d) s_pos[d] = s_lvl[d];
    for (int j = 0; j < NR; ++j) {         // counting-sort nodes by level
      int d = depth[j]; if (d > maxd) d = maxd;
      orderg[s_pos[d]++] = j;
    }
  }
  __syncthreads();
  const int nlev = s_nlev;

  for (int t = 1; t < NT; ++t) {
    const float* qp = q_prime + (size_t)(t - 1) * NR;  // scan over rows 0..T-2
    const float* qn = q_prime + (size_t)t * NR;
    // phase 1: b_i partial = c3*q + c4*clip(q')  ; prefetch next row
    for (int i = tid; i < NR; i += RT_THREADS) {
      float ql = fmaxf(qp[i], Q_LB_C);
      s_acc[i] = c3g[i] * s_q[i] + c4g[i] * ql;
      __builtin_prefetch(qn + i, 0, 1);                // global_prefetch_b8
    }
    __syncthreads();
    // phase 2: + c2[p] * inflow (scatter-add over tree edges)
    for (int j = tid; j < NR - 1; j += RT_THREADS) {
      int p = s_ds[j];
      atomicAdd(&s_acc[p], c2g[p] * s_q[j]);           // ds_add_f32
    }
    __syncthreads();
    // phase 3: forward substitution level-by-level (parents at deeper levels)
    for (int d = 0; d < nlev; ++d) {
      int beg = s_lvl[d], end = s_lvl[d + 1];
      for (int idx = beg + tid; idx < end; idx += RT_THREADS) {
        int i = orderg[idx];
        float x = s_acc[i];                // final: all children contributed
        if (i != NR - 1) {
          int p = s_ds[i];
          atomicAdd(&s_acc[p], c1g[p] * x);
        }
        s_q[i] = fmaxf(x, Q_LB_C);         // clip AFTER solve, per reference
      }
      __syncthreads();
    }
    if (tid == 0) out[t] = s_q[NR - 1];
  }
}

// ---------------------------------------------------------------------------
extern "C" void kernel_launch(void* const* d_in, const int* in_sizes, int n_in,
                              void* d_out, int out_size, void* d_ws, size_t ws_size,
                              hipStream_t stream) {
  const float* q_prime = (const float*)d_in[0];   // [730, 4096]
  const float* nman    = (const float*)d_in[1];
  const float* qsp     = (const float*)d_in[2];
  const float* psp     = (const float*)d_in[3];
  const float* len     = (const float*)d_in[4];
  const float* slp     = (const float*)d_in[5];
  const float* wid     = (const float*)d_in[6];
  const float* xst     = (const float*)d_in[7];
  const float* adj     = (const float*)d_in[8];   // [4096, 4096]
  float* out = (float*)d_out;                      // [1, 730] -> 730 floats

  // workspace: ds[4096] i32 | order[4096] i32 | c1..c4[4096] f32  (~96KB)
  int*   ds    = (int*)d_ws;
  int*   order = ds + NR;
  float* c1    = (float*)(ds + 2 * NR);
  float* c2    = c1 + NR;
  float* c3    = c2 + NR;
  float* c4    = c3 + NR;

  mc_coeff_kernel<<<(NR + 255) / 256, 256, 0, stream>>>(
      nman, qsp, psp, len, slp, wid, xst, c1, c2, c3, c4);
  mc_ds_wmma_kernel<<<NR / 16, 32, 0, stream>>>(adj, ds);
  mc_route_kernel<<<1, RT_THREADS, 0, stream>>>(
      q_prime, ds, order, c1, c2, c3, c4, out);
}